// RetinalDiseaseKGEncoder_39264591020094
// MI455X (gfx1250) — compile-verified
//
#include <hip/hip_runtime.h>

typedef float v2f __attribute__((ext_vector_type(2)));
typedef float v4f __attribute__((ext_vector_type(4)));
typedef float v8f __attribute__((ext_vector_type(8)));

#define HID 64
#define NNODES 4096

// ---------------------------------------------------------------------------
// Kernel A: C_j = h(4096x64) @ W_j(64x64).  blockIdx.y = j in [0,9):
//   j<8  -> W_j = w_rel[l][j],  output Bws[j]
//   j==8 -> W_j = w_self[l],    output Sout (partial slot 8)
// LDS holds W in pair-interleaved layout: float2 (K=2p, K=2p+1) at
// lds[(p*80 + n)*2 + {0,1}], so each WMMA B operand is one ds b64 load.
// ---------------------------------------------------------------------------
__global__ __launch_bounds__(256) void kg_weight_gemm(
    const float* __restrict__ h,
    const float* __restrict__ w_self,   // 64x64 (this layer)
    const float* __restrict__ w_rel,    // 8x64x64 (this layer)
    float* __restrict__ Bws,            // 8 x (4096x64)
    float* __restrict__ Sout)           // 4096x64
{
  __shared__ float ldsW[32 * 160];      // 32 K-pairs, stride 80 float2 (20KB)
  const int t = threadIdx.x;
  const int j = blockIdx.y;
  const float* W = (j < 8) ? (w_rel + (size_t)j * (HID * HID)) : w_self;

  // Stage 64x64 weights into LDS (coalesced float4 reads).
  #pragma unroll
  for (int i = 0; i < 4; ++i) {
    int idx = i * 256 + t;              // 0..1023 float4 units
    int k   = idx >> 4;                 // 0..63
    int n4  = (idx & 15) << 2;          // 0..60
    v4f v = *(const v4f*)(W + k * HID + n4);
    int base = ((k >> 1) * 80 + n4) * 2 + (k & 1);
    ldsW[base + 0] = v.x; ldsW[base + 2] = v.y;
    ldsW[base + 4] = v.z; ldsW[base + 6] = v.w;
  }
  __syncthreads();

  const int lane    = t & 31;
  const int wave    = t >> 5;
  const int lane15  = lane & 15;
  const int half    = lane >> 4;                 // K sub-offset select
  const int rowBase = blockIdx.x * 64 + (wave >> 1) * 16;
  const int n0      = (wave & 1) * 32;
  const int row     = rowBase + lane15;
  const float* Arow = h + (size_t)row * HID + 2 * half;

  v8f acc0 = {}; v8f acc1 = {};
  #pragma unroll
  for (int k0 = 0; k0 < HID; k0 += 4) {
    int p = (k0 >> 1) + half;
    v2f a  = *(const v2f*)(Arow + k0);                               // K = k0+2h, +1
    v2f b0 = *(const v2f*)(&ldsW[(p * 80 + n0 + lane15) * 2]);
    v2f b1 = *(const v2f*)(&ldsW[(p * 80 + n0 + 16 + lane15) * 2]);
    acc0 = __builtin_amdgcn_wmma_f32_16x16x4_f32(false, a, false, b0, (short)0, acc0, false, false);
    acc1 = __builtin_amdgcn_wmma_f32_16x16x4_f32(false, a, false, b1, (short)0, acc1, false, false);
  }

  float* Cout = (j < 8) ? (Bws + (size_t)j * (NNODES * HID)) : Sout;
  #pragma unroll
  for (int v = 0; v < 8; ++v) {
    int rr = rowBase + v + 8 * half;             // C/D layout: VGPR v -> M = v + 8*half
    Cout[(size_t)rr * HID + n0 + lane15]      = acc0[v];
    Cout[(size_t)rr * HID + n0 + 16 + lane15] = acc1[v];
  }
}

// ---------------------------------------------------------------------------
// Kernel B: P_r = rel_adj[r] (4096x4096) @ B_r (4096x64).
// grid = (64 row-tiles, 8 relations); 256 threads = 8 waves.
// Wave w: rows (w>>1)*16..+15 of the 64-row tile, columns (w&1)*32..+31.
// K loop in blocks of 128; B block staged in LDS (40KB, pair-interleaved);
// A operand loaded straight from HBM as one b64 per lane (streaming, read once).
// ---------------------------------------------------------------------------
__global__ __launch_bounds__(256) void kg_adj_gemm(
    const float* __restrict__ rel_adj,  // 8 x 4096 x 4096
    const float* __restrict__ Bws,      // 8 x (4096x64)
    float* __restrict__ P)              // 9 slots of 4096x64; writes slot r
{
  __shared__ float ldsB[64 * 160];      // 64 K-pairs, stride 80 float2 (40KB)
  const int t = threadIdx.x;
  const int r = blockIdx.y;
  const float* Bsrc = Bws + (size_t)r * (NNODES * HID);

  const int lane    = t & 31;
  const int wave    = t >> 5;
  const int lane15  = lane & 15;
  const int half    = lane >> 4;
  const int rowBase = blockIdx.x * 64 + (wave >> 1) * 16;
  const int n0      = (wave & 1) * 32;
  const int row     = rowBase + lane15;
  const float* Arow = rel_adj + ((size_t)r << 24) + (size_t)row * NNODES + 2 * half;

  v8f acc0 = {}; v8f acc1 = {};
  for (int kb = 0; kb < NNODES; kb += 128) {
    __syncthreads();
    #pragma unroll
    for (int i = 0; i < 8; ++i) {
      int idx = i * 256 + t;            // 0..2047 float4 units
      int k   = idx >> 4;               // 0..127
      int n4  = (idx & 15) << 2;
      v4f v = *(const v4f*)(Bsrc + (size_t)(kb + k) * HID + n4);
      int base = ((k >> 1) * 80 + n4) * 2 + (k & 1);
      ldsB[base + 0] = v.x; ldsB[base + 2] = v.y;
      ldsB[base + 4] = v.z; ldsB[base + 6] = v.w;
    }
    __syncthreads();
    #pragma unroll
    for (int k0 = 0; k0 < 128; k0 += 4) {
      int p = (k0 >> 1) + half;
      v2f a  = *(const v2f*)(Arow + kb + k0);
      v2f b0 = *(const v2f*)(&ldsB[(p * 80 + n0 + lane15) * 2]);
      v2f b1 = *(const v2f*)(&ldsB[(p * 80 + n0 + 16 + lane15) * 2]);
      acc0 = __builtin_amdgcn_wmma_f32_16x16x4_f32(false, a, false, b0, (short)0, acc0, false, false);
      acc1 = __builtin_amdgcn_wmma_f32_16x16x4_f32(false, a, false, b1, (short)0, acc1, false, false);
    }
  }

  float* Cout = P + (size_t)r * (NNODES * HID);
  #pragma unroll
  for (int v = 0; v < 8; ++v) {
    int rr = rowBase + v + 8 * half;
    Cout[(size_t)rr * HID + n0 + lane15]      = acc0[v];
    Cout[(size_t)rr * HID + n0 + 16 + lane15] = acc1[v];
  }
}

// h_out = relu( P[8] (=S) + sum_{r<8} P[r] ), elementwise over 4096x64.
__global__ __launch_bounds__(256) void kg_reduce_relu(
    const float* __restrict__ P, float* __restrict__ hout)
{
  int idx = (blockIdx.x * 256 + threadIdx.x) * 4;
  v4f s = *(const v4f*)(P + (size_t)8 * (NNODES * HID) + idx);
  #pragma unroll
  for (int j = 0; j < 8; ++j)
    s += *(const v4f*)(P + (size_t)j * (NNODES * HID) + idx);
  s.x = fmaxf(s.x, 0.0f); s.y = fmaxf(s.y, 0.0f);
  s.z = fmaxf(s.z, 0.0f); s.w = fmaxf(s.w, 0.0f);
  *(v4f*)(hout + idx) = s;
}

// out[kw, d] = (id < 4096) ? h[id, d] : 0   (id == 4096 is the zero pad row)
__global__ __launch_bounds__(256) void kg_gather(
    const float* __restrict__ h, const int* __restrict__ ids,
    float* __restrict__ out)
{
  int tid = blockIdx.x * 256 + threadIdx.x;   // 0..131071
  int kw = tid >> 6;
  int d  = tid & 63;
  int id = ids[kw];
  out[tid] = ((unsigned)id < (unsigned)NNODES) ? h[(size_t)id * HID + d] : 0.0f;
}

extern "C" void kernel_launch(void* const* d_in, const int* in_sizes, int n_in,
                              void* d_out, int out_size, void* d_ws, size_t ws_size,
                              hipStream_t stream) {
  const float* node_embed = (const float*)d_in[0];   // 4097 x 64
  const float* w_self     = (const float*)d_in[1];   // 2 x 64 x 64
  const float* w_rel      = (const float*)d_in[2];   // 2 x 8 x 64 x 64
  const float* rel_adj    = (const float*)d_in[3];   // 8 x 4096 x 4096
  const int*   ids        = (const int*)d_in[4];     // 64 x 32
  float* out = (float*)d_out;

  float* ws = (float*)d_ws;
  const size_t HN = (size_t)NNODES * HID;            // 262144 floats
  float* h1  = ws;                                   // layer-1 activations
  float* h2  = ws + HN;                              // layer-2 activations
  float* Bws = ws + 2 * HN;                          // 8 panels  (8 MB)
  float* P   = ws + 10 * HN;                         // 9 partial slots (9 MB)

  const float* hin = node_embed;                     // rows 0..4095 of embed
  for (int l = 0; l < 2; ++l) {
    kg_weight_gemm<<<dim3(64, 9), 256, 0, stream>>>(
        hin,
        w_self + (size_t)l * HID * HID,
        w_rel  + (size_t)l * 8 * HID * HID,
        Bws, P + (size_t)8 * HN);
    kg_adj_gemm<<<dim3(64, 8), 256, 0, stream>>>(rel_adj, Bws, P);
    float* hout = (l == 0) ? h1 : h2;
    kg_reduce_relu<<<256, 256, 0, stream>>>(P, hout);
    hin = hout;
  }
  kg_gather<<<131072 / 256, 256, 0, stream>>>(h2, ids, out);
}